// EdgeDecoder_9405978378812
// MI455X (gfx1250) — compile-verified
//
#include <hip/hip_runtime.h>

typedef __attribute__((ext_vector_type(2))) float v2f;
typedef __attribute__((ext_vector_type(4))) float v4f;
typedef __attribute__((ext_vector_type(8))) float v8f;

#define HIDDEN 128
#define W1COLS 256

static __device__ __forceinline__ v8f wmma_f32(v2f a, v2f b, v8f c) {
    return __builtin_amdgcn_wmma_f32_16x16x4_f32(
        /*neg_a=*/false, a, /*neg_b=*/false, b,
        /*c_mod=*/(short)0, c, /*reuse_a=*/false, /*reuse_b=*/false);
}

// ---------------------------------------------------------------------------
// Projection GEMM: out[r][n] = sum_{k<128} emb[r][k] * w1[n][k_off + k]
// w1 is [128 x 256] row-major.  One wave computes a 32x16 tile (two 16x16
// WMMA tiles sharing one B tile => 3 loads feed 2 independent v_wmma).
// 8 waves/block cover 32 rows x 128 cols; grid = ceil(rows/32).
// ---------------------------------------------------------------------------
__global__ __launch_bounds__(256) void proj_f32_wmma_kernel(
    const float* __restrict__ emb, const float* __restrict__ w1,
    int k_off, float* __restrict__ out, int rows)
{
    const int wave = threadIdx.x >> 5;          // 0..7 -> column tile
    const int lane = threadIdx.x & 31;
    const int row0 = blockIdx.x * 32;
    const int colT = wave * 16;

    const int mn  = lane & 15;                  // A row in tile == B col in tile
    const int khi = (lane >> 4) * 2;            // K sub-offset: 0 or 2

    int ar0 = row0 + mn;       if (ar0 >= rows) ar0 = rows - 1;   // clamp loads
    int ar1 = row0 + 16 + mn;  if (ar1 >= rows) ar1 = rows - 1;
    const float* a0p = emb + (size_t)ar0 * HIDDEN + khi;
    const float* a1p = emb + (size_t)ar1 * HIDDEN + khi;
    // B[k][n] = w1[colT+n][k_off + k] -> contiguous float2 per lane
    const float* bp  = w1 + (size_t)(colT + mn) * W1COLS + k_off + khi;

    v8f acc0 = {}, acc1 = {};
#pragma unroll 4
    for (int k = 0; k < HIDDEN; k += 4) {
        v2f b  = *(const v2f*)(bp  + k);
        v2f a0 = *(const v2f*)(a0p + k);
        v2f a1 = *(const v2f*)(a1p + k);
        acc0 = wmma_f32(a0, b, acc0);
        acc1 = wmma_f32(a1, b, acc1);
    }

    // D layout: vgpr v holds row (v + 8*(lane>=16)), col = lane&15
    const int rsub = (lane >> 4) << 3;
    const int col  = colT + (lane & 15);

    float* o0 = out + (size_t)(row0 + rsub) * HIDDEN + col;
    if (row0 + 16 <= rows) {                    // uniform: full tile
#pragma unroll
        for (int v = 0; v < 8; ++v) o0[(size_t)v * HIDDEN] = acc0[v];
    } else {                                    // tail block only
#pragma unroll
        for (int v = 0; v < 8; ++v)
            if (row0 + rsub + v < rows) o0[(size_t)v * HIDDEN] = acc0[v];
    }
    float* o1 = out + (size_t)(row0 + 16 + rsub) * HIDDEN + col;
    if (row0 + 32 <= rows) {
#pragma unroll
        for (int v = 0; v < 8; ++v) o1[(size_t)v * HIDDEN] = acc1[v];
    } else {
#pragma unroll
        for (int v = 0; v < 8; ++v)
            if (row0 + 16 + rsub + v < rows) o1[(size_t)v * HIDDEN] = acc1[v];
    }
}

// ---------------------------------------------------------------------------
// Edge phase: out[e] = relu(pu[i] + pm[j] + b1) . w2 + b2
// One wave32 per edge; lane handles 4 of 128 features (one b128 load each).
// pu/pm gathers are L2 hits (76.8 MB resident in 192 MB L2); the index
// stream and output are single-use -> non-temporal.
// ---------------------------------------------------------------------------
__global__ __launch_bounds__(256) void edge_combine_kernel(
    const float* __restrict__ pu, const float* __restrict__ pm,
    const long long* __restrict__ eidx, long long nE,
    const float* __restrict__ b1, const float* __restrict__ w2,
    const float* __restrict__ b2, float* __restrict__ out)
{
    const int wave = threadIdx.x >> 5;
    const int lane = threadIdx.x & 31;
    const long long e = (long long)blockIdx.x * 8 + wave;
    if (e >= nE) return;

    const long long ui = __builtin_nontemporal_load(eidx + e);
    const long long mi = __builtin_nontemporal_load(eidx + nE + e);
    const int f = lane * 4;

    v4f a  = *(const v4f*)(pu + ui * HIDDEN + f);
    v4f b  = *(const v4f*)(pm + mi * HIDDEN + f);
    v4f bb = *(const v4f*)(b1 + f);
    v4f w  = *(const v4f*)(w2 + f);

    float p = 0.f;
#pragma unroll
    for (int t = 0; t < 4; ++t) {
        float h = a[t] + b[t] + bb[t];
        h = h > 0.f ? h : 0.f;
        p = fmaf(h, w[t], p);
    }
#pragma unroll
    for (int off = 16; off > 0; off >>= 1)
        p += __shfl_xor(p, off, 32);

    if (lane == 0) __builtin_nontemporal_store(p + b2[0], out + e);
}

// ---------------------------------------------------------------------------
// Fallback (only if workspace too small): direct per-edge MLP.
// One wave per edge; x staged in LDS; lane computes 4 hidden features.
// ---------------------------------------------------------------------------
__global__ __launch_bounds__(256) void edge_direct_kernel(
    const float* __restrict__ ue, const float* __restrict__ me,
    const long long* __restrict__ eidx, long long nE,
    const float* __restrict__ w1, const float* __restrict__ b1,
    const float* __restrict__ w2, const float* __restrict__ b2,
    float* __restrict__ out)
{
    __shared__ float xs[8][2 * HIDDEN];
    const int wave = threadIdx.x >> 5;
    const int lane = threadIdx.x & 31;
    const long long e = (long long)blockIdx.x * 8 + wave;
    if (e >= nE) return;

    const long long ui = eidx[e];
    const long long mi = eidx[nE + e];
    *(v4f*)&xs[wave][lane * 4]          = *(const v4f*)(ue + ui * HIDDEN + lane * 4);
    *(v4f*)&xs[wave][HIDDEN + lane * 4] = *(const v4f*)(me + mi * HIDDEN + lane * 4);
    // wave-synchronous: same wave wrote this LDS slice; compiler orders via DScnt

    float acc = 0.f;
#pragma unroll
    for (int t = 0; t < 4; ++t) {
        const int n = lane + 32 * t;
        float s = b1[n];
        const float* wr = w1 + (size_t)n * W1COLS;
#pragma unroll 8
        for (int k = 0; k < 2 * HIDDEN; ++k)
            s = fmaf(xs[wave][k], wr[k], s);
        s = s > 0.f ? s : 0.f;
        acc = fmaf(s, w2[n], acc);
    }
#pragma unroll
    for (int off = 16; off > 0; off >>= 1)
        acc += __shfl_xor(acc, off, 32);
    if (lane == 0) out[e] = acc + b2[0];
}

extern "C" void kernel_launch(void* const* d_in, const int* in_sizes, int n_in,
                              void* d_out, int out_size, void* d_ws, size_t ws_size,
                              hipStream_t stream) {
    const float*     user_emb  = (const float*)d_in[0];
    const float*     movie_emb = (const float*)d_in[1];
    const long long* eidx      = (const long long*)d_in[2];
    const float*     w1        = (const float*)d_in[3];
    const float*     b1        = (const float*)d_in[4];
    const float*     w2        = (const float*)d_in[5];
    const float*     b2        = (const float*)d_in[6];
    float*           out       = (float*)d_out;

    const long long nU = (long long)in_sizes[0] / HIDDEN;
    const long long nM = (long long)in_sizes[1] / HIDDEN;
    const long long nE = (long long)in_sizes[2] / 2;

    const size_t need = (size_t)(nU + nM) * HIDDEN * sizeof(float);
    const int gridE = (int)((nE + 7) / 8);

    if (ws_size >= need) {
        float* pu = (float*)d_ws;
        float* pm = pu + (size_t)nU * HIDDEN;
        const int gU = (int)((nU + 31) / 32);
        const int gM = (int)((nM + 31) / 32);
        proj_f32_wmma_kernel<<<gU, 256, 0, stream>>>(user_emb, w1, 0,       pu, (int)nU);
        proj_f32_wmma_kernel<<<gM, 256, 0, stream>>>(movie_emb, w1, HIDDEN, pm, (int)nM);
        edge_combine_kernel<<<gridE, 256, 0, stream>>>(pu, pm, eidx, nE, b1, w2, b2, out);
    } else {
        edge_direct_kernel<<<gridE, 256, 0, stream>>>(user_emb, movie_emb, eidx, nE,
                                                      w1, b1, w2, b2, out);
    }
}